// TrajFM_8194797601379
// MI455X (gfx1250) — compile-verified
//
#include <hip/hip_runtime.h>
#include <hip/hip_bf16.h>
#include <math.h>

typedef __attribute__((ext_vector_type(2))) float v2f;
typedef __attribute__((ext_vector_type(8))) float v8f;

#define NTOK 4096
#define E_ 128
#define D_ 256
#define NPOI_ 10000
#define HD_ 32

#define NT 8              // 16-wide N tiles per wave (wave tile 16x128)
#define AS_STRIDE 20      // even (aligned b64 pairs), mult of 4 (b128 staging), 20m%64 distinct
#define BS_STRIDE 288     // floats per k-pair row in Bs (256 data + 32 pad -> bank shift 32)

__device__ __forceinline__ float leaky(float v) { return v >= 0.f ? v : 0.01f * v; }

// ---- CDNA5 async global->LDS copies (ASYNCcnt path) ----
__device__ __forceinline__ unsigned lds_off(const void* p) {
  return (unsigned)(uintptr_t)(__attribute__((address_space(3))) const void*)p;
}
__device__ __forceinline__ void async_ld_b32(float* l, const float* g) {
  asm volatile("global_load_async_to_lds_b32 %0, %1, off"
               :: "v"(lds_off(l)), "v"((unsigned long long)(uintptr_t)g)
               : "memory");
}
__device__ __forceinline__ void async_ld_b128(float* l, const float* g) {
  asm volatile("global_load_async_to_lds_b128 %0, %1, off"
               :: "v"(lds_off(l)), "v"((unsigned long long)(uintptr_t)g)
               : "memory");
}
__device__ __forceinline__ void wait_async0() {
  asm volatile("s_wait_asynccnt 0x0" ::: "memory");
}

// ---------------- WMMA fp32 GEMM:  C[M,N] = act(A[M,K] @ B[K,N] + bias) ----------------
// block = 128 threads (4 waves); block tile 64x128; wave tile 16x128; K-step 16.
// Double-buffered LDS, tiles staged with global_load_async_to_lds (ASYNCcnt).
// A fragment:  lanes 0-15 hold K{kk,kk+1}, lanes 16-31 hold K{kk+2,kk+3}; M = lane%16.
// B fragment:  mirrored; N = lane%16 (+16*t). B staged pair-interleaved so each lane's
// (K,K+1) pair is one aligned ds_load_b64.
__global__ __launch_bounds__(128) void wmma_gemm_f32(
    const float* __restrict__ A, const float* __restrict__ B,
    const float* __restrict__ bias, float* __restrict__ C,
    int M, int N, int K, int lda, int ldb, int ldc, int act)
{
  (void)M; (void)N;
  __shared__ __align__(16) float As[2][64 * AS_STRIDE];   // [m][k]
  __shared__ __align__(16) float Bs[2][8 * BS_STRIDE];    // [kpair][n*2 + k&1]
  const int tid  = threadIdx.x;
  const int wave = tid >> 5;
  const int lane = tid & 31;
  const int lidx = lane & 15;            // M for A-frag, N for B-frag
  const int khalf = (lane >> 4) << 1;    // 0 or 2
  const int m_blk = blockIdx.y * 64;
  const int n_blk = blockIdx.x * 128;

  v8f acc[NT];
  const v8f zero = {0.f,0.f,0.f,0.f,0.f,0.f,0.f,0.f};
#pragma unroll
  for (int t = 0; t < NT; ++t) acc[t] = zero;

  auto stage = [&](int buf, int k0) {
    // A tile 64x16: 256 x b128 (2 per thread), coalesced
#pragma unroll
    for (int idx = tid; idx < 256; idx += 128) {
      int r = idx >> 2, c4 = (idx & 3) << 2;
      async_ld_b128(&As[buf][r * AS_STRIDE + c4],
                    &A[(size_t)(m_blk + r) * lda + k0 + c4]);
    }
    // B tile 16x128, pair-interleaved scatter: b32 (16 per thread), coalesced on global side
#pragma unroll
    for (int idx = tid; idx < 16 * 128; idx += 128) {
      int r = idx >> 7, c = idx & 127;
      async_ld_b32(&Bs[buf][(r >> 1) * BS_STRIDE + c * 2 + (r & 1)],
                   &B[(size_t)(k0 + r) * ldb + n_blk + c]);
    }
  };

  stage(0, 0);
  wait_async0();
  __syncthreads();

  int cur = 0;
  for (int k0 = 0; k0 < K; k0 += 16) {
    int nbuf = cur ^ 1;
    if (k0 + 16 < K) stage(nbuf, k0 + 16);   // prefetch next tile while computing
    const float* as = &As[cur][0];
    const float* bs = &Bs[cur][0];
#pragma unroll
    for (int kk = 0; kk < 16; kk += 4) {
      v2f a = *(const v2f*)&as[(wave * 16 + lidx) * AS_STRIDE + kk + khalf]; // even -> aligned b64
      const int bbase = ((kk + khalf) >> 1) * BS_STRIDE + lidx * 2;
      // batch all 8 B-fragment loads first so the scheduler keeps them in flight
      v2f bf[NT];
#pragma unroll
      for (int t = 0; t < NT; ++t)
        bf[t] = *(const v2f*)&bs[bbase + t * 32];                            // aligned b64
#pragma unroll
      for (int t = 0; t < NT; ++t)
        acc[t] = __builtin_amdgcn_wmma_f32_16x16x4_f32(
            false, a, false, bf[t], (short)0, acc[t], false, false);
    }
    wait_async0();
    __syncthreads();
    cur = nbuf;
  }

  const int row_half = (lane >> 4) * 8;
#pragma unroll
  for (int t = 0; t < NT; ++t) {
#pragma unroll
    for (int r = 0; r < 8; ++r) {
      int gm = m_blk + wave * 16 + row_half + r;
      int gn = n_blk + t * 16 + lidx;
      float v = acc[t][r];
      if (bias) v += bias[gn];
      if (act == 1) v = fmaxf(v, 0.f);
      C[(size_t)gm * ldc + gn] = v;
    }
  }
}

// ---------------- per-token prep: token-LN rows, spatial hidden, fourier time feats ----------------
__global__ __launch_bounds__(128) void k_prep(
    const float* __restrict__ input_seq,
    const float* __restrict__ tok_emb, const float* __restrict__ tok_g, const float* __restrict__ tok_b,
    const float* __restrict__ sp_w1, const float* __restrict__ sp_b1,
    const float* __restrict__ four_w, const float* __restrict__ four_b,
    float* __restrict__ tokln, float* __restrict__ hidden1, float* __restrict__ tfeat)
{
  int n = blockIdx.x, tid = threadIdx.x;
  const float* iq = input_seq + (size_t)n * 8;   // (4 comps, 2 chans)
  float sx = iq[0], sy = iq[2], ts = iq[4], dtv = iq[6];
  int t0 = (int)iq[1], t1 = (int)iq[5];
  __shared__ float red[128];
  for (int w = 0; w < 2; ++w) {
    int tk = (w == 0) ? t0 : t1;
    float v = tok_emb[tk * E_ + tid];
    red[tid] = v; __syncthreads();
    for (int s = 64; s > 0; s >>= 1) { if (tid < s) red[tid] += red[tid + s]; __syncthreads(); }
    float mean = red[0] * (1.f / 128.f); __syncthreads();
    float dd = v - mean; red[tid] = dd * dd; __syncthreads();
    for (int s = 64; s > 0; s >>= 1) { if (tid < s) red[tid] += red[tid + s]; __syncthreads(); }
    float var = red[0] * (1.f / 128.f); __syncthreads();
    tokln[(size_t)(2 * n + w) * E_ + tid] = dd * rsqrtf(var + 1e-5f) * tok_g[tid] + tok_b[tid];
  }
  hidden1[(size_t)n * E_ + tid] = leaky(sx * sp_w1[tid] + sy * sp_w1[E_ + tid] + sp_b1[tid]);
  float tt[4];
  tt[0] = fmodf(ts, 604800.f) * (1.f / 86400.f);
  tt[1] = fmodf(ts, 86400.f)  * (1.f / 3600.f);
  tt[2] = fmodf(ts, 3600.f)   * (1.f / 60.f);
  tt[3] = dtv * (1.f / 60.f);
#pragma unroll
  for (int f = 0; f < 4; ++f)
    tfeat[(size_t)n * 512 + f * E_ + tid] =
        leaky(cosf(tt[f] * four_w[f * E_ + tid] + four_b[f * E_ + tid]));
}

// ---------------- POI argmin + gather + LN ----------------
__global__ __launch_bounds__(128) void k_poi(
    const float* __restrict__ input_seq, const float* __restrict__ poi_coors,
    const float* __restrict__ poi_embed, const float* __restrict__ g, const float* __restrict__ b,
    float* __restrict__ poiln)
{
  int n = blockIdx.x, tid = threadIdx.x;
  float sx = input_seq[(size_t)n * 8 + 0];
  float sy = input_seq[(size_t)n * 8 + 2];
  float best = 3.4e38f; int bi = NPOI_;
  for (int p = tid; p < NPOI_; p += 128) {
    float dx = poi_coors[2 * p] - sx, dy = poi_coors[2 * p + 1] - sy;
    float d2 = dx * dx + dy * dy;
    if (d2 < best) { best = d2; bi = p; }   // strided ascending -> first occurrence per thread
  }
  __shared__ float sd[128]; __shared__ int si[128];
  sd[tid] = best; si[tid] = bi; __syncthreads();
  for (int s = 64; s > 0; s >>= 1) {
    if (tid < s) {
      if (sd[tid + s] < sd[tid] || (sd[tid + s] == sd[tid] && si[tid + s] < si[tid])) {
        sd[tid] = sd[tid + s]; si[tid] = si[tid + s];
      }
    }
    __syncthreads();
  }
  int poi = si[0]; __syncthreads();
  __shared__ float red[128];
  float v = poi_embed[(size_t)poi * E_ + tid];
  red[tid] = v; __syncthreads();
  for (int s = 64; s > 0; s >>= 1) { if (tid < s) red[tid] += red[tid + s]; __syncthreads(); }
  float mean = red[0] * (1.f / 128.f); __syncthreads();
  float dd = v - mean; red[tid] = dd * dd; __syncthreads();
  for (int s = 64; s > 0; s >>= 1) { if (tid < s) red[tid] += red[tid + s]; __syncthreads(); }
  float var = red[0] * (1.f / 128.f);
  poiln[(size_t)n * E_ + tid] = dd * rsqrtf(var + 1e-5f) * g[tid] + b[tid];
}

// ---------------- add tok_e into x per modality ----------------
__global__ __launch_bounds__(256) void k_combine(const float* __restrict__ tok_e, float* __restrict__ x)
{
  int idx = blockIdx.x * 256 + threadIdx.x;   // 12288*256 total
  int d = idx & 255;
  int row = idx >> 8;                          // n*3 + mod
  int n = row / 3, mod = row - n * 3;
  int tr = 2 * n + (mod == 1 ? 1 : 0);
  x[idx] += tok_e[(size_t)tr * D_ + d];
}

// ---------------- 3-wide per-head attention ----------------
__global__ __launch_bounds__(128) void k_attn(const float* __restrict__ qkv, float* __restrict__ o)
{
  int idx = blockIdx.x * 128 + threadIdx.x;    // NTOK*3*8 = 98304 threads
  int n = idx / 24; int r = idx % 24; int i = r >> 3; int h = r & 7;
  const float* qp = qkv + (size_t)(n * 3 + i) * 768 + h * HD_;
  float s[3];
#pragma unroll
  for (int j = 0; j < 3; ++j) {
    const float* kp = qkv + (size_t)(n * 3 + j) * 768 + 256 + h * HD_;
    float acc = 0.f;
#pragma unroll
    for (int d = 0; d < HD_; ++d) acc += qp[d] * kp[d];
    s[j] = acc * 0.17677669529663687f;         // 1/sqrt(32)
  }
  float m = fmaxf(s[0], fmaxf(s[1], s[2]));
  float e0 = expf(s[0] - m), e1 = expf(s[1] - m), e2 = expf(s[2] - m);
  float inv = 1.f / (e0 + e1 + e2);
  const float* v0 = qkv + (size_t)(n * 3 + 0) * 768 + 512 + h * HD_;
  const float* v1 = qkv + (size_t)(n * 3 + 1) * 768 + 512 + h * HD_;
  const float* v2 = qkv + (size_t)(n * 3 + 2) * 768 + 512 + h * HD_;
  float* op = o + (size_t)(n * 3 + i) * D_ + h * HD_;
#pragma unroll
  for (int d = 0; d < HD_; ++d)
    op[d] = (e0 * v0[d] + e1 * v1[d] + e2 * v2[d]) * inv;
}

// ---------------- out = LN(a + b) over D=256 ----------------
__global__ __launch_bounds__(256) void k_add_ln(
    const float* __restrict__ a, const float* __restrict__ b,
    const float* __restrict__ g, const float* __restrict__ beta,
    float* __restrict__ out)
{
  int row = blockIdx.x, tid = threadIdx.x;
  float v = a[(size_t)row * D_ + tid] + b[(size_t)row * D_ + tid];
  __shared__ float red[256];
  red[tid] = v; __syncthreads();
  for (int s = 128; s > 0; s >>= 1) { if (tid < s) red[tid] += red[tid + s]; __syncthreads(); }
  float mean = red[0] * (1.f / 256.f); __syncthreads();
  float dd = v - mean; red[tid] = dd * dd; __syncthreads();
  for (int s = 128; s > 0; s >>= 1) { if (tid < s) red[tid] += red[tid + s]; __syncthreads(); }
  float var = red[0] * (1.f / 256.f);
  out[(size_t)row * D_ + tid] = dd * rsqrtf(var + 1e-5f) * g[tid] + beta[tid];
}

// ---------------- modality mean + positional encodings ----------------
__global__ __launch_bounds__(256) void k_final(
    const float* __restrict__ x2, const int* __restrict__ positions, float* __restrict__ out)
{
  int n = blockIdx.x, d = threadIdx.x;
  float m = (x2[(size_t)(n * 3 + 0) * D_ + d] +
             x2[(size_t)(n * 3 + 1) * D_ + d] +
             x2[(size_t)(n * 3 + 2) * D_ + d]) * (1.f / 3.f);
  int p0 = positions[2 * n], p1 = positions[2 * n + 1];
  int i = d >> 1;
  float freq = expf(-logf(10000.f) * (2.f * (float)i) / 256.f);
  float a0 = (float)p0 * freq, a1 = (float)p1 * freq;
  float pe = (d & 1) ? (cosf(a0) + cosf(a1)) : (sinf(a0) + sinf(a1));
  out[(size_t)n * D_ + d] = m + pe;
}

extern "C" void kernel_launch(void* const* d_in, const int* in_sizes, int n_in,
                              void* d_out, int out_size, void* d_ws, size_t ws_size,
                              hipStream_t stream) {
  (void)in_sizes; (void)n_in; (void)out_size; (void)ws_size;
  const float* input_seq = (const float*)d_in[0];
  const int*   positions = (const int*)d_in[1];
  const float* poi_coors = (const float*)d_in[2];
  const float* poi_embed = (const float*)d_in[3];
  const float* sp_w1 = (const float*)d_in[4];
  const float* sp_b1 = (const float*)d_in[5];
  const float* sp_w2 = (const float*)d_in[6];
  const float* sp_b2 = (const float*)d_in[7];
  const float* four_w = (const float*)d_in[8];
  const float* four_b = (const float*)d_in[9];
  const float* te_w = (const float*)d_in[10];
  const float* te_b = (const float*)d_in[11];
  const float* poi_ln_g = (const float*)d_in[12];
  const float* poi_ln_b = (const float*)d_in[13];
  const float* poi_w = (const float*)d_in[14];
  const float* poi_b = (const float*)d_in[15];
  const float* tok_emb = (const float*)d_in[16];
  const float* tok_ln_g = (const float*)d_in[17];
  const float* tok_ln_b = (const float*)d_in[18];
  const float* tok_w = (const float*)d_in[19];
  const float* tok_b = (const float*)d_in[20];
  const float* mha_in_w = (const float*)d_in[21];
  const float* mha_in_b = (const float*)d_in[22];
  const float* mha_out_w = (const float*)d_in[23];
  const float* mha_out_b = (const float*)d_in[24];
  const float* ln1_g = (const float*)d_in[25];
  const float* ln1_b = (const float*)d_in[26];
  const float* ff_w1 = (const float*)d_in[27];
  const float* ff_b1 = (const float*)d_in[28];
  const float* ff_w2 = (const float*)d_in[29];
  const float* ff_b2 = (const float*)d_in[30];
  const float* ln2_g = (const float*)d_in[31];
  const float* ln2_b = (const float*)d_in[32];
  float* out = (float*)d_out;

  // workspace layout (floats)
  float* ws = (float*)d_ws;
  float* tokln   = ws;                           // 8192 x 128
  float* hidden1 = tokln + (size_t)8192 * 128;   // 4096 x 128
  float* tfeat   = hidden1 + (size_t)4096 * 128; // 4096 x 512
  float* poiln   = tfeat + (size_t)4096 * 512;   // 4096 x 128
  float* tok_e   = poiln + (size_t)4096 * 128;   // 8192 x 256
  float* x       = tok_e + (size_t)8192 * 256;   // 12288 x 256
  float* qkv     = x + (size_t)12288 * 256;      // 12288 x 768
  float* obuf    = qkv + (size_t)12288 * 768;    // 12288 x 256
  float* t1      = obuf + (size_t)12288 * 256;   // 12288 x 256
  float* t2      = t1 + (size_t)12288 * 256;     // 12288 x 256

  auto gemm = [&](const float* A, const float* B, const float* bias, float* C,
                  int M, int N, int K, int lda, int ldb, int ldc, int act) {
    dim3 g(N / 128, M / 64), blk(128);
    wmma_gemm_f32<<<g, blk, 0, stream>>>(A, B, bias, C, M, N, K, lda, ldb, ldc, act);
  };

  k_prep<<<NTOK, 128, 0, stream>>>(input_seq, tok_emb, tok_ln_g, tok_ln_b,
                                   sp_w1, sp_b1, four_w, four_b,
                                   tokln, hidden1, tfeat);
  k_poi<<<NTOK, 128, 0, stream>>>(input_seq, poi_coors, poi_embed, poi_ln_g, poi_ln_b, poiln);

  // embedding projections (x rows are n*3+mod; view as token rows of 768)
  gemm(tokln,   tok_w, tok_b, tok_e,   8192, 256, 128, 128, 256, 256, 0);
  gemm(hidden1, sp_w2, sp_b2, x + 0,   4096, 256, 128, 128, 256, 768, 0);
  gemm(tfeat,   te_w,  te_b,  x + 256, 4096, 256, 512, 512, 256, 768, 0);
  gemm(poiln,   poi_w, poi_b, x + 512, 4096, 256, 128, 128, 256, 768, 0);
  k_combine<<<12288, 256, 0, stream>>>(tok_e, x);

  // attention block
  gemm(x, mha_in_w, mha_in_b, qkv, 12288, 768, 256, 256, 768, 768, 0);
  k_attn<<<768, 128, 0, stream>>>(qkv, obuf);
  gemm(obuf, mha_out_w, mha_out_b, t1, 12288, 256, 256, 256, 256, 256, 0);
  k_add_ln<<<12288, 256, 0, stream>>>(x, t1, ln1_g, ln1_b, t2);   // t2 = x1

  // FFN block
  gemm(t2, ff_w1, ff_b1, t1, 12288, 256, 256, 256, 256, 256, 1);  // relu
  gemm(t1, ff_w2, ff_b2, obuf, 12288, 256, 256, 256, 256, 256, 0);
  k_add_ln<<<12288, 256, 0, stream>>>(t2, obuf, ln2_g, ln2_b, t1); // t1 = x2

  k_final<<<NTOK, 256, 0, stream>>>(t1, positions, out);
}